// RoutingGNN_60670708023801
// MI455X (gfx1250) — compile-verified
//
#include <hip/hip_runtime.h>
#include <hip/hip_bf16.h>

// ---------------------------------------------------------------------------
// RoutingGNN (2-layer GCN) for MI455X / gfx1250.
//
// Roofline: GEMMs ~0.26 GFLOP (noise). Dominant cost: 2 rounds of
// gather(256MB) + scatter-add(256MB) over 2M edges x 32 f32 feats -> ~1GB,
// ~45us at 23.3 TB/s; acc buffers (12.8MB) live in the 192MB L2 so the
// f32 atomics resolve in-cache. Everything stays fp32 (matches reference);
// the dense per-node GEMMs use V_WMMA_F32_16X16X4_F32 (fp32 WMMA, wave32).
//
// Norm factorization: g = (x@W) * dis[node];  agg[d] = dis[d]*(sum_e g[src] + g[d])
// => self-loop handled elementwise, only one dis factor applied per side.
// ---------------------------------------------------------------------------

typedef __attribute__((ext_vector_type(2))) float v2f;
typedef __attribute__((ext_vector_type(8))) float v8f;

__global__ void gcn_fill_deg(float* __restrict__ deg, int n) {
    int i = blockIdx.x * blockDim.x + threadIdx.x;
    if (i < n) deg[i] = 1.0f;                       // self-loop contributes 1
}

__global__ void gcn_count_deg(const long long* __restrict__ dst,
                              float* __restrict__ deg, int nE) {
    int e = blockIdx.x * blockDim.x + threadIdx.x;
    if (e < nE) atomicAdd(&deg[(int)dst[e]], 1.0f); // exact: deg < 2^24
}

__global__ void gcn_rsqrt(float* __restrict__ dis, int n) {
    int i = blockIdx.x * blockDim.x + threadIdx.x;
    if (i < n) dis[i] = rsqrtf(dis[i]);             // deg >= 1 always (self-loop)
}

// One wave per 16-node tile. D = A(16xK) * W(Kx32) as two 16x16 col tiles,
// K stepped by 4 with V_WMMA_F32_16X16X4_F32. Writes G = D * dis[row] and
// zeroes Acc rows (scatter targets) for the following edge-scatter kernel.
// Fragment layout (ISA 7.12.2, wave32):
//   A : lane l(+16h) vgpr v -> A[tile*16+l][k0+2h+v]
//   B : lane l(+16h) vgpr v -> W[k0+2h+v][col16+l]
//   C : lane l(+16h) vgpr v -> D[v+8h][col16+l]
template<int K>
__global__ void gcn_gemm_wmma(const float* __restrict__ A,
                              const float* __restrict__ W,
                              const float* __restrict__ dis,
                              float* __restrict__ G,
                              float* __restrict__ Acc,
                              int ntiles) {
    int wave = blockIdx.x * (blockDim.x >> 5) + (threadIdx.x >> 5);
    if (wave >= ntiles) return;                     // wave-uniform: EXEC stays all-1s
    const int lane = threadIdx.x & 31;
    const int h    = lane >> 4;                     // K-half select
    const int l    = lane & 15;
    const int m    = wave * 16 + l;                 // A row this lane feeds
    v8f c0 = {}; v8f c1 = {};
#pragma unroll
    for (int ks = 0; ks < K / 4; ++ks) {
        const int k = ks * 4 + 2 * h;
        v2f a, b0, b1;
        a.x  = A[m * K + k];
        a.y  = A[m * K + k + 1];
        b0.x = W[k * 32 + l];
        b0.y = W[(k + 1) * 32 + l];
        b1.x = W[k * 32 + 16 + l];
        b1.y = W[(k + 1) * 32 + 16 + l];
        // (neg_a, A, neg_b, B, c_mod, C, reuse_a, reuse_b)
        c0 = __builtin_amdgcn_wmma_f32_16x16x4_f32(false, a, false, b0,
                                                   (short)0, c0, false, false);
        c1 = __builtin_amdgcn_wmma_f32_16x16x4_f32(false, a, false, b1,
                                                   (short)0, c1, false, false);
    }
#pragma unroll
    for (int v = 0; v < 8; ++v) {
        const int   r  = wave * 16 + v + 8 * h;
        const float dr = dis[r];
        G[r * 32 + l]        = c0[v] * dr;          // g = (A@W) * dis
        G[r * 32 + 16 + l]   = c1[v] * dr;
        Acc[r * 32 + l]      = 0.0f;                // zero scatter targets
        Acc[r * 32 + 16 + l] = 0.0f;
    }
}

// Wave per edge, lane per feature: 128B coalesced gather of g[src], 128B of
// coalesced global_atomic_add_f32 into acc[dst].
__global__ void gcn_scatter(const long long* __restrict__ src,
                            const long long* __restrict__ dst,
                            const float* __restrict__ g,
                            float* __restrict__ acc, int nE) {
    long long tid = (long long)blockIdx.x * blockDim.x + threadIdx.x;
    int e = (int)(tid >> 5);
    if (e >= nE) return;
    int f = (int)(tid & 31);
    int s = (int)src[e];
    int d = (int)dst[e];
    atomicAdd(&acc[d * 32 + f], g[s * 32 + f]);
}

// out[i,f] = relu(dis[i] * (acc[i,f] + g[i,f]) + bias[f])   (g term = self loop)
__global__ void gcn_relu_bias(const float* __restrict__ acc,
                              const float* __restrict__ g,
                              const float* __restrict__ dis,
                              const float* __restrict__ bias,
                              float* __restrict__ out, int n) {
    long long tid = (long long)blockIdx.x * blockDim.x + threadIdx.x;
    int i = (int)(tid >> 5);
    if (i >= n) return;
    int f = (int)(tid & 31);
    float v = dis[i] * (acc[i * 32 + f] + g[i * 32 + f]) + bias[f];
    out[i * 32 + f] = v > 0.0f ? v : 0.0f;
}

// Final head: y[i] = relu(dis*(acc+g)+b2) . Wf + bf   (Fout = 1)
__global__ void gcn_final(const float* __restrict__ acc,
                          const float* __restrict__ g,
                          const float* __restrict__ dis,
                          const float* __restrict__ b2,
                          const float* __restrict__ Wf,
                          const float* __restrict__ bf,
                          float* __restrict__ out, int n) {
    int i = blockIdx.x * blockDim.x + threadIdx.x;
    if (i >= n) return;
    const float di = dis[i];
    float s = bf[0];
#pragma unroll
    for (int f = 0; f < 32; ++f) {
        float h = di * (acc[i * 32 + f] + g[i * 32 + f]) + b2[f];
        h = h > 0.0f ? h : 0.0f;
        s += h * Wf[f];
    }
    out[i] = s;
}

extern "C" void kernel_launch(void* const* d_in, const int* in_sizes, int n_in,
                              void* d_out, int out_size, void* d_ws, size_t ws_size,
                              hipStream_t stream) {
    const float*     x  = (const float*)d_in[0];      // [N,8]
    const long long* ei = (const long long*)d_in[1];  // [2,E] int64 row-major
    const float*     W1 = (const float*)d_in[2];      // [8,32]
    const float*     b1 = (const float*)d_in[3];      // [32]
    const float*     W2 = (const float*)d_in[4];      // [32,32]
    const float*     b2 = (const float*)d_in[5];      // [32]
    const float*     Wf = (const float*)d_in[6];      // [32,1]
    const float*     bf = (const float*)d_in[7];      // [1]
    float* out = (float*)d_out;

    const int N = in_sizes[0] / 8;
    const int E = in_sizes[1] / 2;
    const long long* src = ei;        // edge_index[0]
    const long long* dst = ei + E;    // edge_index[1]

    // Workspace partition (dis + g + acc + h) ~ 38.8 MB for N=100k.
    char*  ws  = (char*)d_ws;
    size_t off = 0;
    auto  carve = [&](size_t bytes) -> float* {
        float* p = (float*)(ws + off);
        off = (off + bytes + 255) & ~(size_t)255;
        return p;
    };
    float* dis  = carve((size_t)N * sizeof(float));        // deg -> dis (in place)
    float* bufG = carve((size_t)N * 32 * sizeof(float));   // g = (h@W)*dis
    float* bufA = carve((size_t)N * 32 * sizeof(float));   // scatter accumulator
    float* bufH = carve((size_t)N * 32 * sizeof(float));   // relu activations
    (void)ws_size; (void)n_in; (void)out_size;

    const int ntiles = (N + 15) / 16;                      // 100000/16 = 6250 exact

    dim3 blk(256);
    // 1) degree (self-loops baked in as init=1)
    gcn_fill_deg<<<dim3((N + 255) / 256), blk, 0, stream>>>(dis, N);
    gcn_count_deg<<<dim3((E + 255) / 256), blk, 0, stream>>>(dst, dis, E);
    gcn_rsqrt<<<dim3((N + 255) / 256), blk, 0, stream>>>(dis, N);

    // 2) layer 1: g1 = (x@W1)*dis ; acc = 0
    gcn_gemm_wmma<8><<<dim3((ntiles + 7) / 8), blk, 0, stream>>>(x, W1, dis, bufG, bufA, ntiles);
    gcn_scatter<<<dim3((int)(((long long)E * 32 + 255) / 256)), blk, 0, stream>>>(src, dst, bufG, bufA, E);
    gcn_relu_bias<<<dim3((int)(((long long)N * 32 + 255) / 256)), blk, 0, stream>>>(bufA, bufG, dis, b1, bufH, N);

    // 3) layer 2: g2 = (h@W2)*dis ; acc = 0
    gcn_gemm_wmma<32><<<dim3((ntiles + 7) / 8), blk, 0, stream>>>(bufH, W2, dis, bufG, bufA, ntiles);
    gcn_scatter<<<dim3((int)(((long long)E * 32 + 255) / 256)), blk, 0, stream>>>(src, dst, bufG, bufA, E);

    // 4) head
    gcn_final<<<dim3((N + 255) / 256), blk, 0, stream>>>(bufA, bufG, dis, b2, Wf, bf, out, N);
}